// DGATNet_17867063951387
// MI455X (gfx1250) — compile-verified
//
#include <hip/hip_runtime.h>

// ---------------------------------------------------------------------------
// DGATNet DiffusioUnpool — sparse D(S+I)D x formulation for MI455X (gfx1250)
// wave32; CSR built on device; gather = 4 edges x 8 float4 feature-quads per
// wave with CSR edge chunks staged into LDS via GLOBAL_LOAD_ASYNC_TO_LDS.
// ---------------------------------------------------------------------------

#if defined(__has_builtin)
#if __has_builtin(__builtin_amdgcn_global_load_async_to_lds_b32) && \
    __has_builtin(__builtin_amdgcn_s_wait_asynccnt)
#define HAVE_ASYNC_LDS 1
#endif
#endif

#ifndef HAVE_ASYNC_LDS
#define HAVE_ASYNC_LDS 0
#endif

#define GAS __attribute__((address_space(1)))
#define LAS __attribute__((address_space(3)))

typedef int iv4 __attribute__((ext_vector_type(4)));
typedef float fv4 __attribute__((ext_vector_type(4)));

__global__ void k_init(float* __restrict__ deg, float* __restrict__ y_a,
                       float* __restrict__ y_f, int* __restrict__ cnt,
                       int* __restrict__ cursor, int Nn, int n_yf) {
  int i = blockIdx.x * blockDim.x + threadIdx.x;
  int nq = n_yf >> 2;
  if (i < nq) {
    fv4 z = {0.0f, 0.0f, 0.0f, 0.0f};
    ((fv4*)y_f)[i] = z;  // b128 zero fill
  }
  if (i == nq) {  // scalar tail (n_yf % 4)
    for (int k = nq * 4; k < n_yf; ++k) y_f[k] = 0.0f;
  }
  if (i < Nn) {
    deg[i] = 1.0f;  // self loop from A + I
    y_a[i] = 0.0f;
    cnt[i] = 0;
    cursor[i] = 0;
  }
}

// quad-edge degree accumulation (b128 NT streaming loads)
__global__ void k_degree4(const int* __restrict__ erow,
                          const float* __restrict__ eattr,
                          float* __restrict__ deg, int* __restrict__ cnt,
                          int Ne) {
  int base = (blockIdx.x * blockDim.x + threadIdx.x) * 4;
  if (base >= Ne) return;
  iv4 r4 = __builtin_nontemporal_load((const iv4*)(erow + base));
  fv4 a4 = __builtin_nontemporal_load((const fv4*)(eattr + base));
#pragma unroll
  for (int k = 0; k < 4; ++k) {
    atomicAdd(&deg[r4[k]], a4[k]);
    atomicAdd(&cnt[r4[k]], 1);
  }
}

__global__ void k_degree(const int* __restrict__ erow,
                         const float* __restrict__ eattr,
                         float* __restrict__ deg, int* __restrict__ cnt,
                         int Ne) {
  int e = blockIdx.x * blockDim.x + threadIdx.x;
  if (e >= Ne) return;
  int r = erow[e];
  atomicAdd(&deg[r], eattr[e]);
  atomicAdd(&cnt[r], 1);
}

// single workgroup (1024 threads) exclusive prefix sum over Nn row counts
__global__ void k_scan(const int* __restrict__ cnt, int* __restrict__ rowptr,
                       int Nn) {
  __shared__ int sums[1024];
  int t = threadIdx.x;
  int per = (Nn + 1023) / 1024;
  int beg = t * per;
  int end = beg + per;
  if (end > Nn) end = Nn;
  int s = 0;
  for (int i = beg; i < end; ++i) s += cnt[i];
  sums[t] = s;
  __syncthreads();
  for (int off = 1; off < 1024; off <<= 1) {  // Hillis-Steele inclusive scan
    int v = (t >= off) ? sums[t - off] : 0;
    __syncthreads();
    sums[t] += v;
    __syncthreads();
  }
  int base = (t == 0) ? 0 : sums[t - 1];
  for (int i = beg; i < end; ++i) {
    rowptr[i] = base;
    base += cnt[i];
  }
  if (t == 1023) rowptr[Nn] = sums[1023];
}

__global__ void k_rsqrt(float* __restrict__ deg, int Nn) {
  int i = blockIdx.x * blockDim.x + threadIdx.x;
  if (i < Nn) deg[i] = rsqrtf(deg[i]);  // deg -> d = deg^{-1/2} in place
}

// fdim==32 fast path: float4 per thread (b128 NT in, b128 out)
__global__ void k_scatter_y32(const float* __restrict__ fea,
                              const int* __restrict__ perm,
                              const float* __restrict__ coffe,
                              const float* __restrict__ d,
                              float* __restrict__ y_f, float* __restrict__ y_a,
                              const int* __restrict__ bs_ptr, int n_perm,
                              int n_coffe) {
  int t = blockIdx.x * blockDim.x + threadIdx.x;
  if (t >= n_perm * 8) return;
  int j = t >> 3;
  int fq = t & 7;
  int p = perm[j];
  float dp = d[p];
  fv4 v = __builtin_nontemporal_load((const fv4*)fea + t);
  fv4 o = {dp * v.x, dp * v.y, dp * v.z, dp * v.w};
  ((fv4*)y_f)[p * 8 + fq] = o;  // reused many times by gather: keep RT
  if (fq == 0) {
    int batch = bs_ptr[0];                      // device-side scalar input
    int node_num = n_coffe / batch;             // 192
    int per_b = (n_perm / n_coffe) * node_num;  // 768
    int ci = (j / per_b) * node_num + (j % node_num);
    y_a[p] = dp * coffe[ci];
  }
}

__global__ void k_scatter_y_generic(const float* __restrict__ fea,
                                    const int* __restrict__ perm,
                                    const float* __restrict__ coffe,
                                    const float* __restrict__ d,
                                    float* __restrict__ y_f,
                                    float* __restrict__ y_a,
                                    const int* __restrict__ bs_ptr, int n_perm,
                                    int n_coffe, int fdim) {
  int t = blockIdx.x * blockDim.x + threadIdx.x;
  if (t >= n_perm * fdim) return;
  int j = t / fdim;
  int f = t - j * fdim;
  int p = perm[j];
  float dp = d[p];
  y_f[p * fdim + f] = dp * fea[t];
  if (f == 0) {
    int batch = bs_ptr[0];
    int node_num = n_coffe / batch;
    int per_b = (n_perm / n_coffe) * node_num;
    int ci = (j / per_b) * node_num + (j % node_num);
    y_a[p] = dp * coffe[ci];
  }
}

// quad-edge CSR bucket scatter (b128 NT streaming loads)
__global__ void k_csr4(const int* __restrict__ erow,
                       const int* __restrict__ ecol,
                       const float* __restrict__ eattr,
                       const int* __restrict__ rowptr, int* __restrict__ cursor,
                       int* __restrict__ csr_col, float* __restrict__ csr_val,
                       int Ne) {
  int base = (blockIdx.x * blockDim.x + threadIdx.x) * 4;
  if (base >= Ne) return;
  iv4 r4 = __builtin_nontemporal_load((const iv4*)(erow + base));
  iv4 c4 = __builtin_nontemporal_load((const iv4*)(ecol + base));
  fv4 a4 = __builtin_nontemporal_load((const fv4*)(eattr + base));
#pragma unroll
  for (int k = 0; k < 4; ++k) {
    int r = r4[k];
    int slot = rowptr[r] + atomicAdd(&cursor[r], 1);
    csr_col[slot] = c4[k];  // read back by gather soon: keep RT
    csr_val[slot] = a4[k];
  }
}

__global__ void k_csr(const int* __restrict__ erow,
                      const int* __restrict__ ecol,
                      const float* __restrict__ eattr,
                      const int* __restrict__ rowptr, int* __restrict__ cursor,
                      int* __restrict__ csr_col, float* __restrict__ csr_val,
                      int Ne) {
  int e = blockIdx.x * blockDim.x + threadIdx.x;
  if (e >= Ne) return;
  int r = erow[e];
  int slot = rowptr[r] + atomicAdd(&cursor[r], 1);
  csr_col[slot] = ecol[e];
  csr_val[slot] = eattr[e];
}

// ---------------------------------------------------------------------------
// Gather, fdim==32: one wave32 per row. Lane layout: eq = lane>>3 picks one of
// 4 concurrent edges, fq = lane&7 picks a float4 feature quad. Edge chunks
// (32 cols + 32 vals) staged to LDS via async DMA (ASYNCcnt) when available.
// ---------------------------------------------------------------------------
__global__ __launch_bounds__(256) void k_gather32(
    const int* __restrict__ rowptr, const int* __restrict__ csr_col,
    const float* __restrict__ csr_val, const float* __restrict__ y_f,
    const float* __restrict__ y_a, const float* __restrict__ d,
    float* __restrict__ out, int Nn) {
  __shared__ int s_cols[8][32];
  __shared__ float s_vals[8][32];
  const int lane = threadIdx.x & 31;
  const int wslot = threadIdx.x >> 5;
  const int row = (blockIdx.x * blockDim.x + threadIdx.x) >> 5;
  if (row >= Nn) return;
  const int eq = lane >> 3;
  const int fq = lane & 7;

  const int beg = rowptr[row];
  const int end = rowptr[row + 1];

  fv4 acc = {0.0f, 0.0f, 0.0f, 0.0f};
  if (eq == 0)  // + I self-loop term (counted once)
    acc = *(const fv4*)(y_f + row * 32 + fq * 4);
  float acc_a = (lane == 0) ? y_a[row] : 0.0f;

  for (int base = beg; base < end; base += 32) {
    int m = end - base;
    if (m > 32) m = 32;
    int e = base + lane;
    int e_ld = (e < end) ? e : (end - 1);  // clamp: async reads stay in-bounds

#if HAVE_ASYNC_LDS
    {
      int* gpc = const_cast<int*>(csr_col + e_ld);
      int* gpv = reinterpret_cast<int*>(const_cast<float*>(csr_val + e_ld));
      int* lpc = &s_cols[wslot][lane];
      int* lpv = reinterpret_cast<int*>(&s_vals[wslot][lane]);
      __builtin_amdgcn_global_load_async_to_lds_b32((GAS int*)gpc,
                                                    (LAS int*)lpc, 0, 0);
      __builtin_amdgcn_global_load_async_to_lds_b32((GAS int*)gpv,
                                                    (LAS int*)lpv, 0, 0);
    }
#else
    s_cols[wslot][lane] = csr_col[e_ld];
    s_vals[wslot][lane] = csr_val[e_ld];
#endif
    if (base + 32 < end) {  // CDNA5 global_prefetch_b8 for the next chunk
      __builtin_prefetch(&csr_col[base + 32], 0, 0);
      __builtin_prefetch(&csr_val[base + 32], 0, 0);
    }
#if HAVE_ASYNC_LDS
    __builtin_amdgcn_s_wait_asynccnt(0);
    __builtin_amdgcn_fence(__ATOMIC_ACQUIRE, "wavefront");
#endif
    __builtin_amdgcn_wave_barrier();

    // attention channel: one edge per lane
    if (lane < m) {
      int ca = s_cols[wslot][lane];
      float va = s_vals[wslot][lane];
      acc_a = fmaf(va, y_a[ca], acc_a);
    }
    // feature channel: 4 edges per iteration, float4 (b128) per lane
    for (int j = 0; j < m; j += 4) {
      int src = j + eq;
      bool ok = src < m;
      int idx = ok ? src : 0;
      int bc = s_cols[wslot][idx];  // LDS broadcast (8-lane groups)
      float bv = ok ? s_vals[wslot][idx] : 0.0f;
      const fv4 yv = *(const fv4*)(y_f + bc * 32 + fq * 4);
      acc.x = fmaf(bv, yv.x, acc.x);
      acc.y = fmaf(bv, yv.y, acc.y);
      acc.z = fmaf(bv, yv.z, acc.z);
      acc.w = fmaf(bv, yv.w, acc.w);
    }
  }

  // reduce the 4 edge-groups (lanes fq, fq+8, fq+16, fq+24)
  for (int off = 8; off <= 16; off <<= 1) {
    acc.x += __shfl_xor(acc.x, off, 32);
    acc.y += __shfl_xor(acc.y, off, 32);
    acc.z += __shfl_xor(acc.z, off, 32);
    acc.w += __shfl_xor(acc.w, off, 32);
  }
  for (int off = 16; off > 0; off >>= 1) acc_a += __shfl_down(acc_a, off, 32);

  float dr = d[row];
  if (eq == 0) {  // lanes 0..7 emit the 32 features as 8 x b128 NT stores
    fv4 o = {dr * acc.x, dr * acc.y, dr * acc.z, dr * acc.w};
    __builtin_nontemporal_store(o, (fv4*)(out + row * 32 + fq * 4));
  }
  if (lane == 0)  // atte_updata
    __builtin_nontemporal_store(dr * acc_a, out + Nn * 32 + row);
}

// generic fallback (fdim <= 32): lane = feature, shuffle-broadcast edges
__global__ void k_gather_generic(const int* __restrict__ rowptr,
                                 const int* __restrict__ csr_col,
                                 const float* __restrict__ csr_val,
                                 const float* __restrict__ y_f,
                                 const float* __restrict__ y_a,
                                 const float* __restrict__ d,
                                 float* __restrict__ out, int Nn, int fdim) {
  int lane = threadIdx.x & 31;
  int row = (blockIdx.x * blockDim.x + threadIdx.x) >> 5;
  if (row >= Nn) return;
  int beg = rowptr[row];
  int end = rowptr[row + 1];
  float acc = (lane < fdim) ? y_f[row * fdim + lane] : 0.0f;
  float acc_a = (lane == 0) ? y_a[row] : 0.0f;
  for (int base = beg; base < end; base += 32) {
    int e = base + lane;
    int c = 0;
    float v = 0.0f;
    if (e < end) {
      c = csr_col[e];
      v = csr_val[e];
      acc_a = fmaf(v, y_a[c], acc_a);
    }
    int m = end - base;
    if (m > 32) m = 32;
    for (int j = 0; j < m; ++j) {
      int bc = __shfl(c, j, 32);
      float bv = __shfl(v, j, 32);
      if (lane < fdim) acc = fmaf(bv, y_f[bc * fdim + lane], acc);
    }
  }
  for (int off = 16; off > 0; off >>= 1) acc_a += __shfl_down(acc_a, off, 32);
  float dr = d[row];
  if (lane < fdim) out[row * fdim + lane] = dr * acc;
  if (lane == 0) out[Nn * fdim + row] = dr * acc_a;
}

extern "C" void kernel_launch(void* const* d_in, const int* in_sizes, int n_in,
                              void* d_out, int out_size, void* d_ws,
                              size_t ws_size, hipStream_t stream) {
  const float* fea = (const float*)d_in[0];
  const int* perm = (const int*)d_in[1];
  const int* erow = (const int*)d_in[2];
  const float* eattr = (const float*)d_in[3];
  const float* coffe = (const float*)d_in[4];
  const int* bs_ptr = (const int*)d_in[6];

  const int n_perm = in_sizes[1];         // 6144
  const int fdim = in_sizes[0] / n_perm;  // 32
  const int Ne = in_sizes[3];             // 393216
  const int n_coffe = in_sizes[4];        // 1536
  const int Nn = out_size / (fdim + 1);   // 12288
  const int* ecol = erow + Ne;

  // workspace layout (all 4B types; y_f 16B aligned: offset 2*Nn floats)
  float* deg = (float*)d_ws;                 // Nn (becomes d in place)
  float* y_a = deg + Nn;                     // Nn
  float* y_f = y_a + Nn;                     // Nn*fdim
  float* csr_val = y_f + (size_t)Nn * fdim;  // Ne
  int* cnt = (int*)(csr_val + Ne);           // Nn
  int* rowptr = cnt + Nn;                    // Nn+1
  int* cursor = rowptr + Nn + 1;             // Nn
  int* csr_col = cursor + Nn;                // Ne

  const int T = 256;
  const int n_yf = Nn * fdim;
  const bool quad_ok = (Ne % 4) == 0;  // 16B alignment for ecol/eattr quads

  int init_n = (n_yf / 4 + 1) > Nn ? (n_yf / 4 + 1) : Nn;
  hipLaunchKernelGGL(k_init, dim3((init_n + T - 1) / T), dim3(T), 0, stream,
                     deg, y_a, y_f, cnt, cursor, Nn, n_yf);
  if (quad_ok) {
    hipLaunchKernelGGL(k_degree4, dim3((Ne / 4 + T - 1) / T), dim3(T), 0,
                       stream, erow, eattr, deg, cnt, Ne);
  } else {
    hipLaunchKernelGGL(k_degree, dim3((Ne + T - 1) / T), dim3(T), 0, stream,
                       erow, eattr, deg, cnt, Ne);
  }
  hipLaunchKernelGGL(k_scan, dim3(1), dim3(1024), 0, stream, cnt, rowptr, Nn);
  hipLaunchKernelGGL(k_rsqrt, dim3((Nn + T - 1) / T), dim3(T), 0, stream, deg,
                     Nn);
  if (fdim == 32) {
    hipLaunchKernelGGL(k_scatter_y32, dim3((n_perm * 8 + T - 1) / T), dim3(T),
                       0, stream, fea, perm, coffe, deg, y_f, y_a, bs_ptr,
                       n_perm, n_coffe);
  } else {
    hipLaunchKernelGGL(k_scatter_y_generic, dim3((n_perm * fdim + T - 1) / T),
                       dim3(T), 0, stream, fea, perm, coffe, deg, y_f, y_a,
                       bs_ptr, n_perm, n_coffe, fdim);
  }
  if (quad_ok) {
    hipLaunchKernelGGL(k_csr4, dim3((Ne / 4 + T - 1) / T), dim3(T), 0, stream,
                       erow, ecol, eattr, rowptr, cursor, csr_col, csr_val, Ne);
  } else {
    hipLaunchKernelGGL(k_csr, dim3((Ne + T - 1) / T), dim3(T), 0, stream, erow,
                       ecol, eattr, rowptr, cursor, csr_col, csr_val, Ne);
  }
  if (fdim == 32) {
    hipLaunchKernelGGL(k_gather32, dim3((Nn * 32 + T - 1) / T), dim3(T), 0,
                       stream, rowptr, csr_col, csr_val, y_f, y_a, deg,
                       (float*)d_out, Nn);
  } else {
    hipLaunchKernelGGL(k_gather_generic, dim3((Nn * 32 + T - 1) / T), dim3(T),
                       0, stream, rowptr, csr_col, csr_val, y_f, y_a, deg,
                       (float*)d_out, Nn, fdim);
  }
}